// ProposalLayer_81398220193822
// MI455X (gfx1250) — compile-verified
//
#include <hip/hip_runtime.h>
#include <stdint.h>

// ---------------- problem constants ----------------
#define B_      8
#define N_      200000
#define C_      10
#define TOPK_   1000
#define PROP_   200
#define CAND_   4096
#define NBINS   65536
#define TILE_PTS       256
#define TILES_PER_BLK  8
#define SPAN_  (TILE_PTS * TILES_PER_BLK)

typedef unsigned int u32x4 __attribute__((ext_vector_type(4)));
typedef int          i32x4 __attribute__((ext_vector_type(4)));
typedef int          i32x8 __attribute__((ext_vector_type(8)));

// -----------------------------------------------------------------------------
// TDM: 1-D contiguous copy of `ndwords` dwords from global `gaddr` to LDS
// offset `lds_off`.  D# per CDNA5 ISA ch.8:
//   group0: [1:0]=count=1, [63:32]=lds_addr, [120:64]=global_addr, [127:126]=2
//   group1: [17:16]=data_size(2 => 4B), tensor_dim0=nd @ [79:48],
//           tensor_dim1=1 @ [111:80], tile_dim0=nd @ [127:112],
//           tile_dim1/2=0 (unused), tensor_dim0_stride=nd @ [207:160]
// groups 2/3 zero (tile_dim3 unused).  EXEC ignored -> issue from one wave.
// -----------------------------------------------------------------------------
__device__ __forceinline__ void tdm_load_1d(uint32_t lds_off, uint64_t gaddr,
                                            int ndwords) {
  u32x4 g0;
  g0.x = 1u;                                             // count = 1
  g0.y = lds_off;                                        // LDS byte address
  g0.z = (uint32_t)gaddr;                                // global addr [31:0]
  g0.w = ((uint32_t)(gaddr >> 32) & 0x01FFFFFFu)         // global addr [56:32]
         | (2u << 30);                                   // type = 2 ("image")
  i32x8 g1;
  g1[0] = (int)(2u << 16);                               // data_size = 4 bytes
  g1[1] = (int)((uint32_t)ndwords << 16);                // tensor_dim0[15:0]
  g1[2] = (int)(1u << 16);                               // dim0 hi=0, tensor_dim1=1
  g1[3] = (int)((uint32_t)ndwords << 16);                // dim1 hi=0, tile_dim0=nd
  g1[4] = 0;                                             // tile_dim1=0, tile_dim2=0
  g1[5] = ndwords;                                       // tensor_dim0_stride lo32
  g1[6] = 0;
  g1[7] = 0;
  i32x4 gz = {0, 0, 0, 0};
#if defined(__clang__) && (__clang_major__ >= 23)
  i32x8 gz8 = {0, 0, 0, 0, 0, 0, 0, 0};
  __builtin_amdgcn_tensor_load_to_lds(g0, g1, gz, gz, gz8, 0);
#else
  __builtin_amdgcn_tensor_load_to_lds(g0, g1, gz, gz, 0);
#endif
}

// ---------------- K0: zero histogram + candidate counters ----------------
__global__ void k_init(unsigned int* __restrict__ hist,
                       unsigned int* __restrict__ ccnt) {
  size_t g = (size_t)blockIdx.x * blockDim.x + threadIdx.x;
  if (g < (size_t)B_ * NBINS) hist[g] = 0u;
  if (g < B_) ccnt[g] = 0u;
}

// ---------------- K1: streaming score pass (TDM double-buffered) ------------
// For each point: maxlogit + label (first-max, matches jnp.argmax), dir label,
// pack sortable key, histogram scorekey>>16.
__global__ __launch_bounds__(256) void k_score(
    const float* __restrict__ cls, const float* __restrict__ dirp,
    unsigned long long* __restrict__ keys, unsigned int* __restrict__ hist) {
  __shared__ float s_cls[2][TILE_PTS * C_];
  const int b  = blockIdx.y;
  const int p0 = blockIdx.x * SPAN_;
  int span = N_ - p0; if (span > SPAN_) span = SPAN_;
  if (span <= 0) return;
  const int ntiles = (span + TILE_PTS - 1) / TILE_PTS;
  const int tid = threadIdx.x;
  const float* gsrc = cls + ((size_t)b * N_ + p0) * C_;

  if (tid < 32) {                                   // wave 0 drives the TDM
    int pts0 = span < TILE_PTS ? span : TILE_PTS;
    tdm_load_1d((uint32_t)(uintptr_t)&s_cls[0][0],
                (uint64_t)(uintptr_t)gsrc, pts0 * C_);
  }
  for (int t = 0; t < ntiles; ++t) {
    if (tid < 32) {
      if (t + 1 < ntiles) {                         // prefetch next tile
        int pts = span - (t + 1) * TILE_PTS;
        if (pts > TILE_PTS) pts = TILE_PTS;
        tdm_load_1d((uint32_t)(uintptr_t)&s_cls[(t + 1) & 1][0],
                    (uint64_t)(uintptr_t)(gsrc + (size_t)(t + 1) * TILE_PTS * C_),
                    pts * C_);
        __builtin_amdgcn_s_wait_tensorcnt(1);       // tile t complete (in-order)
      } else {
        __builtin_amdgcn_s_wait_tensorcnt(0);
      }
    }
    __syncthreads();
    const int lp = t * TILE_PTS + tid;
    if (lp < span) {
      const int p = p0 + lp;
      const float* v = &s_cls[t & 1][tid * C_];
      float mv = v[0]; int lab = 0;
#pragma unroll
      for (int k = 1; k < C_; ++k) { float x = v[k]; if (x > mv) { mv = x; lab = k; } }
      const float* dp = dirp + ((size_t)b * N_ + p) * 2;
      const int dir = dp[1] > dp[0] ? 1 : 0;        // argmax, first-occurrence
      uint32_t fb = __float_as_uint(mv);
      uint32_t sk = (fb & 0x80000000u) ? ~fb : (fb | 0x80000000u); // sortable
      uint32_t inv = (~(uint32_t)p) & 0x3FFFFu;     // tie-break: lower idx first
      unsigned long long key = ((unsigned long long)sk << 32) |
                               ((unsigned long long)inv << 14) |
                               ((unsigned)lab << 10) | ((unsigned)dir << 9);
      keys[(size_t)b * N_ + p] = key;
      atomicAdd(&hist[(size_t)b * NBINS + (sk >> 16)], 1u);
    }
    __syncthreads();                                // buffer reuse fence
  }
}

// ---------------- K2: per-batch threshold bucket (descending scan) ----------
__global__ __launch_bounds__(256) void k_thresh(
    const unsigned int* __restrict__ hist, unsigned int* __restrict__ tbkt) {
  const int b = blockIdx.x;
  const unsigned int* h = hist + (size_t)b * NBINS;
  __shared__ unsigned int s_part[256];
  __shared__ unsigned int s_cum, s_t, s_done;
  const int tid = threadIdx.x;
  if (tid == 0) { s_cum = 0u; s_t = 0u; s_done = 0u; }
  __syncthreads();
  for (int c = 0; c < NBINS / 256; ++c) {
    const int base = NBINS - 256 * (c + 1);
    s_part[tid] = h[base + tid];
    __syncthreads();
    if (tid == 0) {
      unsigned int sum = 0u;
      for (int i = 0; i < 256; ++i) sum += s_part[i];
      if (s_cum + sum >= (unsigned)TOPK_) {
        unsigned int cum = s_cum;
        for (int i = 255; i >= 0; --i) {
          cum += s_part[i];
          if (cum >= (unsigned)TOPK_) { s_t = (unsigned)(base + i); break; }
        }
        s_done = 1u;
      } else {
        s_cum += sum;
      }
    }
    __syncthreads();
    if (s_done) break;                              // uniform after barrier
  }
  if (tid == 0) tbkt[b] = s_t;
}

// ---------------- K3: gather candidates above/at threshold bucket -----------
__global__ void k_gather(const unsigned long long* __restrict__ keys,
                         const unsigned int* __restrict__ tbkt,
                         unsigned int* __restrict__ ccnt,
                         unsigned long long* __restrict__ cand) {
  size_t gid = (size_t)blockIdx.x * blockDim.x + threadIdx.x;
  if (gid >= (size_t)B_ * N_) return;
  const int b = (int)(gid / N_);
  const unsigned long long key = keys[gid];
  if ((unsigned)(key >> 48) >= tbkt[b]) {
    unsigned int pos = atomicAdd(&ccnt[b], 1u);
    if (pos < CAND_) cand[(size_t)b * CAND_ + pos] = key;
  }
}

// ---------------- K4: per-batch sort + standup + greedy NMS + output --------
__global__ __launch_bounds__(1024) void k_nms(
    const unsigned long long* __restrict__ cand,
    const unsigned int* __restrict__ ccnt,
    const float* __restrict__ box, float* __restrict__ out) {
  __shared__ unsigned long long s_keys[CAND_];      // 32 KB
  __shared__ float s_x0[TOPK_], s_y0[TOPK_], s_x1[TOPK_], s_y1[TOPK_], s_ar[TOPK_];
  __shared__ unsigned char s_keep[TOPK_];
  __shared__ int s_sel[PROP_];
  __shared__ int s_cnt;
  const int b = blockIdx.x;
  const int tid = threadIdx.x;
  int cnt = (int)ccnt[b]; if (cnt > CAND_) cnt = CAND_;

  for (int i = tid; i < CAND_; i += 1024)
    s_keys[i] = (i < cnt) ? cand[(size_t)b * CAND_ + i] : 0ull;
  __syncthreads();

  // bitonic sort, descending
  for (int k = 2; k <= CAND_; k <<= 1) {
    for (int j = k >> 1; j > 0; j >>= 1) {
      for (int i = tid; i < CAND_; i += 1024) {
        const int ixj = i ^ j;
        if (ixj > i) {
          unsigned long long a = s_keys[i], c = s_keys[ixj];
          const bool desc = (i & k) == 0;
          if (desc ? (a < c) : (a > c)) { s_keys[i] = c; s_keys[ixj] = a; }
        }
      }
      __syncthreads();
    }
  }

  // standup boxes for top-1000 (closed form == 4-corner min/max)
  const int M = cnt < TOPK_ ? cnt : TOPK_;
  if (tid < TOPK_) {
    if (tid < M) {
      const unsigned long long key = s_keys[tid];
      const int idx = (int)((~(unsigned)(key >> 14)) & 0x3FFFFu);
      const float* bp = box + ((size_t)b * N_ + idx) * 7;
      const float cx = bp[0], cy = bp[1], w = bp[3], l = bp[4], ang = bp[6];
      const float cc = cosf(ang), ss = sinf(ang);
      const float ex = 0.5f * (fabsf(w * cc) + fabsf(l * ss));
      const float ey = 0.5f * (fabsf(w * ss) + fabsf(l * cc));
      s_x0[tid] = cx - ex; s_y0[tid] = cy - ey;
      s_x1[tid] = cx + ex; s_y1[tid] = cy + ey;
      s_ar[tid] = (ex + ex) * (ey + ey);
      s_keep[tid] = 1;
    } else {
      s_x0[tid] = 0.f; s_y0[tid] = 0.f; s_x1[tid] = 0.f; s_y1[tid] = 0.f;
      s_ar[tid] = 0.f; s_keep[tid] = 0;
    }
  }
  __syncthreads();

  // greedy NMS: outer loop sequential, inner j parallel (j == tid)
  for (int i = 0; i < TOPK_; ++i) {
    if (s_keep[i] && tid > i && tid < TOPK_ && s_keep[tid]) {
      const float lx = fmaxf(s_x0[i], s_x0[tid]);
      const float ly = fmaxf(s_y0[i], s_y0[tid]);
      const float rx = fminf(s_x1[i], s_x1[tid]);
      const float ry = fminf(s_y1[i], s_y1[tid]);
      const float iw = fmaxf(rx - lx, 0.f), ih = fmaxf(ry - ly, 0.f);
      const float inter = iw * ih;
      const float iou = inter / (s_ar[i] + s_ar[tid] - inter + 1e-8f);
      if (iou > 0.1f) s_keep[tid] = 0;
    }
    __syncthreads();
  }

  if (tid == 0) {
    int c = 0;
    for (int i = 0; i < TOPK_ && c < PROP_; ++i)
      if (s_keep[i]) s_sel[c++] = i;
    s_cnt = c;
  }
  __syncthreads();

  if (tid < PROP_) {
    float* ob = out + ((size_t)b * PROP_ + tid) * 7;
    float* oi = out + (size_t)B_ * PROP_ * 7 + (size_t)b * PROP_ + tid;
    float* os = oi + (size_t)B_ * PROP_;
    if (tid < s_cnt) {
      const unsigned long long key = s_keys[s_sel[tid]];
      const int idx = (int)((~(unsigned)(key >> 14)) & 0x3FFFFu);
      const int lab = (int)((key >> 10) & 0xFu);
      const float dir = (float)((key >> 9) & 1u);
      const unsigned sk = (unsigned)(key >> 32);
      const unsigned fb = (sk & 0x80000000u) ? (sk & 0x7FFFFFFFu) : ~sk;
      const float logit = __uint_as_float(fb);
      const float score = 1.0f / (1.0f + expf(-logit));
      const float* bp = box + ((size_t)b * N_ + idx) * 7;
      const float period = 3.14159265358979323846f;   // 2*pi / NUM_DIRECTIONS
      const float ang = bp[6];
      const float rot = ang - floorf(ang / period) * period;
      ob[0] = bp[0]; ob[1] = bp[1]; ob[2] = bp[2];
      ob[3] = bp[3]; ob[4] = bp[4]; ob[5] = bp[5];
      ob[6] = rot + period * dir;
      *oi = (float)lab; *os = score;
    } else {
      for (int k = 0; k < 7; ++k) ob[k] = 0.f;
      *oi = 0.f; *os = 0.f;
    }
  }
}

// -----------------------------------------------------------------------------
extern "C" void kernel_launch(void* const* d_in, const int* in_sizes, int n_in,
                              void* d_out, int out_size, void* d_ws, size_t ws_size,
                              hipStream_t stream) {
  (void)in_sizes; (void)n_in; (void)out_size; (void)ws_size;
  // inputs: [0]=num_class (scalar), [1]=box_pred, [2]=cls_pred, [3]=dir_pred
  const float* box  = (const float*)d_in[1];
  const float* cls  = (const float*)d_in[2];
  const float* dirp = (const float*)d_in[3];
  float* out = (float*)d_out;

  // workspace layout (~15.2 MB total)
  unsigned long long* keys = (unsigned long long*)d_ws;           // B*N u64
  unsigned int* hist = (unsigned int*)(keys + (size_t)B_ * N_);   // B*65536 u32
  unsigned int* tbkt = hist + (size_t)B_ * NBINS;                 // B u32
  unsigned int* ccnt = tbkt + B_;                                 // B u32
  unsigned long long* cand = (unsigned long long*)(ccnt + B_);    // B*CAND u64

  k_init<<<((size_t)B_ * NBINS + 1023) / 1024, 1024, 0, stream>>>(hist, ccnt);
  dim3 g1((N_ + SPAN_ - 1) / SPAN_, B_);
  k_score<<<g1, 256, 0, stream>>>(cls, dirp, keys, hist);
  k_thresh<<<B_, 256, 0, stream>>>(hist, tbkt);
  k_gather<<<(unsigned)(((size_t)B_ * N_ + 255) / 256), 256, 0, stream>>>(
      keys, tbkt, ccnt, cand);
  k_nms<<<B_, 1024, 0, stream>>>(cand, ccnt, box, out);
}